// BatchMatchedMSELoss2_84086869721403
// MI455X (gfx1250) — compile-verified
//
#include <hip/hip_runtime.h>
#include <hip/hip_bf16.h>

// ---------------------------------------------------------------------------
// BatchMatchedMSELoss: loss = mean_i min_j  (||x_i||^2 + ||t_j||^2 - 2<x_i,t_j>)/D
// B = 8192, D = 256, fp32.  Cross GEMM (34.4 GFLOP) on V_WMMA_F32_16X16X4_F32.
// v4: B tiles staged in LDS by the Tensor Data Mover (tensor_load_to_lds,
// TENSORcnt, double-buffered), shared by all 8 waves of the block; A strip
// stays register-resident (64 x v2f = 128 VGPRs).  Inner loop: ds_load_b64 +
// WMMA only.  LDS rows padded to 260 floats -> conflict-free b64 reads.
// (6-arg tensor_load_to_lds builtin on this toolchain.)
// ---------------------------------------------------------------------------

typedef __attribute__((ext_vector_type(2))) float v2f;
typedef __attribute__((ext_vector_type(8))) float v8f;
typedef __attribute__((ext_vector_type(4))) unsigned int u32x4;
typedef __attribute__((ext_vector_type(8))) int i32x8;
typedef __attribute__((ext_vector_type(4))) int i32x4;

#define B_ROWS 8192
#define DDIM   256
#define KSTEPS (DDIM / 4)       // 64 WMMA k-steps per 16x16 tile
#define NTILE_PER_WAVE 32       // 32 x 16 cols = 512 columns per block chunk
#define NCHUNKS (B_ROWS / (16 * NTILE_PER_WAVE))   // 16
#define MTILES  (B_ROWS / 16)                      // 512
#define LDS_STRIDE 260          // 256 floats + 4 pad floats per row

// monotonic encoding of float into uint32 so unsigned atomicMin == float min
__device__ __forceinline__ unsigned int fmin_encode(float f) {
    unsigned int u = __float_as_uint(f);
    return (u & 0x80000000u) ? ~u : (u | 0x80000000u);
}
__device__ __forceinline__ float fmin_decode(unsigned int k) {
    unsigned int u = (k & 0x80000000u) ? (k ^ 0x80000000u) : ~k;
    return __uint_as_float(u);
}

// Issue one TDM load: 16 rows x 256 f32 tile of `target` (row-major, stride 256)
// into LDS at byte offset ldsAddr, padding 4 DWORDs after every 256 DWORDs
// (one row) -> LDS row stride 260 floats.
__device__ __forceinline__ void tdm_load_tile(const float* gsrc, unsigned int ldsAddr) {
    unsigned long long ga = (unsigned long long)(size_t)gsrc;
    u32x4 g0;
    g0.x = 1u;                                   // count=1 (valid), user mode
    g0.y = ldsAddr;                              // lds_addr  [63:32]
    g0.z = (unsigned int)(ga & 0xFFFFFFFFu);     // global_addr[95:64]
    g0.w = (unsigned int)((ga >> 32) & 0x01FFFFFFu) | (2u << 30); // addr hi | type=2

    i32x8 g1;
    g1[0] = (2 << 16)        // data_size = 4 bytes
          | (1 << 20)        // pad_enable
          | (7 << 22)        // pad_interval: 256 DWORDs (= one 1KB row)
          | (3 << 25);       // pad_amount:   4 DWORDs (16B)
    g1[1] = (int)(256u << 16);   // tensor_dim0 = 256 (low 16 bits at [63:48])
    g1[2] = (int)(8192u << 16);  // tensor_dim0 hi = 0; tensor_dim1 = 8192 (low16)
    g1[3] = (int)(256u << 16);   // tensor_dim1 hi = 0; tile_dim0 = 256
    g1[4] = 16;                  // tile_dim1 = 16; tile_dim2 = 0
    g1[5] = 256;                 // tensor_dim0_stride = 256 elements
    g1[6] = 0;
    g1[7] = 0;

    i32x4 z4 = {0, 0, 0, 0};
    i32x8 z8 = {0, 0, 0, 0, 0, 0, 0, 0};
    // 6-arg form on this toolchain: (g0, g1, g2, g3, extra, cpol)
    __builtin_amdgcn_tensor_load_to_lds(g0, g1, z4, z4, z8, 0);
}

// ---- kernel 1: init per-row min keys to +max (ws must be re-inited each call)
__global__ void init_keys_kernel(unsigned int* __restrict__ keys) {
    int i = blockIdx.x * blockDim.x + threadIdx.x;
    if (i < B_ROWS) keys[i] = 0xFFFFFFFFu;
}

// ---- kernel 2: squared row norms, one wave32 per row (8 rows per 256-thr block)
__global__ void sqnorm_kernel(const float* __restrict__ x, float* __restrict__ out) {
    int lane = threadIdx.x & 31;
    int wave = threadIdx.x >> 5;
    int row  = blockIdx.x * 8 + wave;
    const float4* p = (const float4*)(x + (size_t)row * DDIM);
    float4 a = p[lane * 2 + 0];
    float4 b = p[lane * 2 + 1];
    float s = a.x*a.x + a.y*a.y + a.z*a.z + a.w*a.w
            + b.x*b.x + b.y*b.y + b.z*b.z + b.w*b.w;
    #pragma unroll
    for (int m = 16; m >= 1; m >>= 1) s += __shfl_xor(s, m, 32);
    if (lane == 0) out[row] = s;
}

// ---- kernel 3: WMMA GEMM with TDM-staged, double-buffered B tiles in LDS.
__global__ void __launch_bounds__(256)
wmma_minmse_kernel(const float* __restrict__ input,
                   const float* __restrict__ target,
                   const float* __restrict__ tg_sq,
                   unsigned int* __restrict__ keys) {
    __shared__ float bbuf[2][16 * LDS_STRIDE];   // 2 x 16.25KB

    const int lane = threadIdx.x & 31;
    const int wave = threadIdx.x >> 5;
    const int l15  = lane & 15;
    const int half = lane >> 4;   // 0: K=0,1 / C rows 0-7   1: K=2,3 / C rows 8-15

    const int chunk  = blockIdx.x & (NCHUNKS - 1);     // shared by the block
    const int tileM  = (blockIdx.x >> 4) * 8 + wave;   // 0..511

    // A operand: row (tileM*16 + l15), float2 at k + 2*half (ISA 16x4 f32 layout)
    const float* aRow = input + (size_t)(tileM * 16 + l15) * DDIM + 2 * half;
    v2f areg[KSTEPS];
    #pragma unroll
    for (int kk = 0; kk < KSTEPS; ++kk)
        areg[kk] = *(const v2f*)(aRow + 4 * kk);

    float rmin[8];
    #pragma unroll
    for (int r = 0; r < 8; ++r) rmin[r] = 3.4e38f;

    const unsigned int ldsBase[2] = {
        (unsigned int)(size_t)&bbuf[0][0],
        (unsigned int)(size_t)&bbuf[1][0]
    };
    const int colBase = chunk * NTILE_PER_WAVE * 16;

    // preload tile 0 into buffer 0 (one TDM op per block, wave 0)
    if (wave == 0) {
        tdm_load_tile(target + (size_t)colBase * DDIM, ldsBase[0]);
        __builtin_amdgcn_s_wait_tensorcnt(0);
    }
    __syncthreads();

    for (int nt = 0; nt < NTILE_PER_WAVE; ++nt) {
        const int col0 = colBase + nt * 16;
        // prefetch next tile into the other buffer (read in the previous iter;
        // the end-of-iteration barrier below made that safe)
        if (wave == 0 && nt + 1 < NTILE_PER_WAVE)
            tdm_load_tile(target + (size_t)(col0 + 16) * DDIM,
                          ldsBase[(nt + 1) & 1]);

        const float* bRow = &bbuf[nt & 1][l15 * LDS_STRIDE + 2 * half];
        const float tsq   = tg_sq[col0 + l15];

        v8f acc0 = {0.f,0.f,0.f,0.f,0.f,0.f,0.f,0.f};
        v8f acc1 = {0.f,0.f,0.f,0.f,0.f,0.f,0.f,0.f};
        #pragma unroll
        for (int kk = 0; kk < KSTEPS; kk += 2) {
            v2f b0 = *(const v2f*)(bRow + 4 * kk);
            v2f b1 = *(const v2f*)(bRow + 4 * kk + 4);
            acc0 = __builtin_amdgcn_wmma_f32_16x16x4_f32(
                       false, areg[kk],     false, b0, (short)0, acc0, false, false);
            acc1 = __builtin_amdgcn_wmma_f32_16x16x4_f32(
                       false, areg[kk + 1], false, b1, (short)0, acc1, false, false);
        }
        v8f acc = acc0 + acc1;

        // lane holds C rows m = r + 8*half, col n = l15:  cand = tsq - 2*cross
        #pragma unroll
        for (int r = 0; r < 8; ++r)
            rmin[r] = fminf(rmin[r], tsq - 2.0f * acc[r]);

        if (wave == 0 && nt + 1 < NTILE_PER_WAVE)
            __builtin_amdgcn_s_wait_tensorcnt(0);
        __syncthreads();
    }

    // min across the 16 lanes of each half (masks 1..8 stay inside the half)
    #pragma unroll
    for (int r = 0; r < 8; ++r) {
        float v = rmin[r];
        v = fminf(v, __shfl_xor(v, 1, 32));
        v = fminf(v, __shfl_xor(v, 2, 32));
        v = fminf(v, __shfl_xor(v, 4, 32));
        v = fminf(v, __shfl_xor(v, 8, 32));
        if (l15 == 0) {
            int row = tileM * 16 + half * 8 + r;
            atomicMin(&keys[row], fmin_encode(v));
        }
    }
}

// ---- kernel 4: loss_i = (in_sq[i] + min_i)/D ; out = mean_i loss_i
__global__ void finalize_kernel(const unsigned int* __restrict__ keys,
                                const float* __restrict__ in_sq,
                                float* __restrict__ out) {
    __shared__ float red[256];
    float s = 0.f;
    for (int i = threadIdx.x; i < B_ROWS; i += 256)
        s += (in_sq[i] + fmin_decode(keys[i])) * (1.0f / DDIM);
    red[threadIdx.x] = s;
    __syncthreads();
    for (int m = 128; m >= 1; m >>= 1) {
        if (threadIdx.x < m) red[threadIdx.x] += red[threadIdx.x + m];
        __syncthreads();
    }
    if (threadIdx.x == 0) out[0] = red[0] * (1.0f / B_ROWS);
}

extern "C" void kernel_launch(void* const* d_in, const int* in_sizes, int n_in,
                              void* d_out, int out_size, void* d_ws, size_t ws_size,
                              hipStream_t stream) {
    const float* input  = (const float*)d_in[0];
    const float* target = (const float*)d_in[1];
    float* out = (float*)d_out;

    // workspace layout: [keys: 8192 u32][in_sq: 8192 f32][tg_sq: 8192 f32]
    unsigned int* keys = (unsigned int*)d_ws;
    float* in_sq = (float*)d_ws + B_ROWS;
    float* tg_sq = (float*)d_ws + 2 * B_ROWS;

    init_keys_kernel<<<B_ROWS / 256, 256, 0, stream>>>(keys);
    sqnorm_kernel<<<B_ROWS / 8, 256, 0, stream>>>(input, in_sq);
    sqnorm_kernel<<<B_ROWS / 8, 256, 0, stream>>>(target, tg_sq);

    // 1024 blocks x 8 waves: all waves of a block share one N chunk via LDS
    wmma_minmse_kernel<<<(MTILES * NCHUNKS) / 8, 256, 0, stream>>>(
        input, target, tg_sq, keys);

    finalize_kernel<<<1, 256, 0, stream>>>(keys, in_sq, out);
}